// FFFLayer_85100482003665
// MI455X (gfx1250) — compile-verified
//
#include <hip/hip_runtime.h>
#include <hip/hip_bf16.h>
#include <math.h>

// ---------------------------------------------------------------------------
// Fast-feedforward (FFF) layer for MI455X / gfx1250.
//
//   TOKENS = 8192, N_IN = N_OUT = 4096, DEPTH = 12, N_NODES = 4095.
//
// Phase 1 (fff_route): 16-token tiles; per tree level compute the 16 logits as
// the diagonal of a 16x16 f32 WMMA product between the token tile and the
// gathered w1 rows (K swept in steps of 4 with V_WMMA_F32_16X16X4_F32).
// Phase 2 (fff_accum): per token, out = sum_i gelu_i * w2s[node_i] as a
// coalesced float4 streaming pass (w2s is L2-resident: 64 MB << 192 MB L2).
// ---------------------------------------------------------------------------

#define N_IN   4096
#define N_OUT  4096
#define DEPTH  12

typedef __attribute__((ext_vector_type(2))) float v2f;
typedef __attribute__((ext_vector_type(8))) float v8f;

__device__ __forceinline__ float gelu_exact(float x) {
    // jax.nn.gelu(approximate=False): x * Phi(x)
    return 0.5f * x * (1.0f + erff(x * 0.70710678118654752f));
}

// ---------------------------------------------------------------------------
// Phase 1: tree routing via f32 WMMA.
// Grid: tokens/16 blocks. Block: 128 threads = 4 wave32s; wave w owns the
// K-slice [w*1024, (w+1)*1024).
//
// WMMA f32 16x16x4 fragment layout (ISA 7.12.2, 32-bit A 16x4 / B 4x16):
//   lane l, l<16 : A holds (M=l,   K=kb+0..1), B holds (N=l,    K=kb+0..1)
//   lane l, l>=16: A holds (M=l-16,K=kb+2..3), B holds (N=l-16, K=kb+2..3)
// so each lane loads one float2 from x and one from the gathered w1 row.
// C/D layout: VGPR r, lane l: (M = r + (l<16 ? 0 : 8), N = l % 16).
// Diagonal element m lives at (r = m&7, lane = m<8 ? m : m+16).
//
// The K loop uses a compile-time trip count (256) so the backend keeps it a
// uniform SALU-counted loop (no divergent exec masking around the WMMAs,
// whose EXEC must be all-ones per ISA 7.12).
// ---------------------------------------------------------------------------
__global__ __launch_bounds__(128) void fff_route(
    const float* __restrict__ x,      // [TOKENS, N_IN]
    const float* __restrict__ w1s,    // [N_NODES, N_IN]
    float*       __restrict__ g_ws,   // [TOKENS, DEPTH]
    int*         __restrict__ node_ws // [TOKENS, DEPTH]
) {
    __shared__ float c_ls[4][8][32];  // [wave][C vgpr][lane] partial products
    __shared__ int   curnode[16];     // current tree node per token in tile

    const int tid   = threadIdx.x;
    const int wave  = tid >> 5;
    const int lane  = tid & 31;
    const int row   = lane & 15;      // token row for A == column n for B
    const int khalf = lane >> 4;      // 0: K pair {0,1}; 1: K pair {2,3}
    const int b0    = blockIdx.x * 16;

    // Per-lane base pointer into this tile's x rows: wave's K-slice start
    // plus the lane's K-pair offset within a 4-wide WMMA step.
    const int kslice = wave * (N_IN / 4);   // 1024 K per wave
    const float* xrow = x + (size_t)(b0 + row) * N_IN + kslice + 2 * khalf;

    if (tid < 16) curnode[tid] = 0;
    __syncthreads();

    for (int level = 0; level < DEPTH; ++level) {
        // Gather this level's w1 row for column n = row (uniform per column).
        const int node = curnode[row];
        const float* w1row =
            w1s + (size_t)node * N_IN + kslice + 2 * khalf;

        v8f c = {};
        #pragma unroll 8
        for (int t = 0; t < (N_IN / 4) / 4; ++t) {   // 256 uniform iterations
            const int kb = 4 * t;
            v2f a = *(const v2f*)(xrow + kb);
            v2f b = *(const v2f*)(w1row + kb);
            // (neg_a, A, neg_b, B, c_mod, C, reuse_a, reuse_b)
            c = __builtin_amdgcn_wmma_f32_16x16x4_f32(
                    false, a, false, b, (short)0, c, false, false);
        }

        #pragma unroll
        for (int r = 0; r < 8; ++r) c_ls[wave][r][lane] = c[r];
        __syncthreads();

        if (tid < 16) {
            const int m  = tid;
            const int r  = m & 7;
            const int ln = (m < 8) ? m : (m + 16);
            const float logit = c_ls[0][r][ln] + c_ls[1][r][ln] +
                                c_ls[2][r][ln] + c_ls[3][r][ln];
            const int b = b0 + m;
            g_ws[b * DEPTH + level]    = gelu_exact(logit);
            node_ws[b * DEPTH + level] = curnode[m];
            curnode[m] = 2 * curnode[m] + 1 + (logit > 0.0f ? 1 : 0);
        }
        __syncthreads();   // publish curnode (and recycle c_ls) for next level
    }
}

// ---------------------------------------------------------------------------
// Phase 2: out[b] = sum_i g[b][i] * w2s[node[b][i]].
// One block per token, 256 threads, 16 floats (4 x float4) per thread,
// float4 chunks interleaved across threads for full coalescing.
// ---------------------------------------------------------------------------
__global__ __launch_bounds__(256) void fff_accum(
    const float* __restrict__ w2s,     // [N_NODES, N_OUT]
    const float* __restrict__ g_ws,    // [TOKENS, DEPTH]
    const int*   __restrict__ node_ws, // [TOKENS, DEPTH]
    float*       __restrict__ out      // [TOKENS, N_OUT]
) {
    const int b   = blockIdx.x;
    const int tid = threadIdx.x;

    float        g[DEPTH];
    const float* w2row[DEPTH];
    #pragma unroll
    for (int i = 0; i < DEPTH; ++i) {
        g[i]     = g_ws[b * DEPTH + i];
        w2row[i] = w2s + (size_t)node_ws[b * DEPTH + i] * N_OUT;
    }

    float* orow = out + (size_t)b * N_OUT;
    #pragma unroll
    for (int j = 0; j < N_OUT / (256 * 4); ++j) {
        const int idx = (tid + 256 * j) * 4;
        float4 acc = make_float4(0.f, 0.f, 0.f, 0.f);
        #pragma unroll
        for (int i = 0; i < DEPTH; ++i) {
            const float4 wv = *(const float4*)(w2row[i] + idx);
            acc.x += g[i] * wv.x;
            acc.y += g[i] * wv.y;
            acc.z += g[i] * wv.z;
            acc.w += g[i] * wv.w;
        }
        *(float4*)(orow + idx) = acc;
    }
}

// ---------------------------------------------------------------------------
// Launcher. Inputs (setup_inputs order): input[8192,4096] f32,
// w1s[4095,4096] f32, w2s[4095,4096] f32, depth (scalar, fixed 12).
// Workspace: g (tokens*12 f32) + nodes (tokens*12 i32) = 768 KB.
// ---------------------------------------------------------------------------
extern "C" void kernel_launch(void* const* d_in, const int* in_sizes, int n_in,
                              void* d_out, int out_size, void* d_ws, size_t ws_size,
                              hipStream_t stream) {
    const float* x   = (const float*)d_in[0];
    const float* w1s = (const float*)d_in[1];
    const float* w2s = (const float*)d_in[2];
    float* out = (float*)d_out;

    const int tokens = in_sizes[0] / N_IN;   // 8192

    float* g_ws    = (float*)d_ws;
    int*   node_ws = (int*)((char*)d_ws + (size_t)tokens * DEPTH * sizeof(float));

    fff_route<<<tokens / 16, 128, 0, stream>>>(x, w1s, g_ws, node_ws);
    fff_accum<<<tokens, 256, 0, stream>>>(w2s, g_ws, node_ws, out);
}